// FrameReducer_76467597738065
// MI455X (gfx1250) — compile-verified
//
#include <hip/hip_runtime.h>
#include <cstdint>
#include <cstddef>

// FrameReducer for MI455X (gfx1250).
// Zero-FLOP compaction/gather: bandwidth-bound (~65 MB moved -> ~3us @ 23.3 TB/s).
// CDNA5 paths used: wave32 ballot stream-compaction, async global<->LDS DMA
// (ASYNCcnt) for the 2KB row copies.

#ifndef __has_builtin
#define __has_builtin(x) 0
#endif

#define LOG_0P9 (-0.10536051565782628f) // log(0.9), rounds to nearest f32

#if defined(__HIP_DEVICE_COMPILE__)
// Typed pointers matching the gfx1250 async builtin signatures:
//   arg0: int4 in global (__device__, AS1), arg1: int4 in LDS (__shared__, AS3)
typedef int v4i __attribute__((ext_vector_type(4)));
typedef __attribute__((address_space(1))) v4i* g_v4i_ptr;
typedef __attribute__((address_space(3))) v4i* l_v4i_ptr;
#endif

__device__ __forceinline__ void wait_async_zero() {
#if defined(__HIP_DEVICE_COMPILE__)
#if __has_builtin(__builtin_amdgcn_s_wait_asynccnt)
    __builtin_amdgcn_s_wait_asynccnt(0);
#else
    asm volatile("s_wait_asynccnt 0x0" ::: "memory");
#endif
#endif
}

__device__ __forceinline__ unsigned ballot32(bool p) {
#if defined(__HIP_DEVICE_COMPILE__) && __has_builtin(__builtin_amdgcn_ballot_w32)
    return __builtin_amdgcn_ballot_w32(p);
#else
    return (unsigned)__ballot(p);
#endif
}

// ---------------------------------------------------------------------------
// Kernel 1: per-batch-row stream compaction of kept frame indices.
// One block = one wave32 per n. Ballot + popc prefix gives a stable compaction
// (kept frames first, original order) exactly matching stable argsort(~mask).
// ---------------------------------------------------------------------------
__global__ void __launch_bounds__(32)
fr_scan_kernel(const float* __restrict__ ctc,      // [N, T, V]
               const int*   __restrict__ x_lens,   // [N]
               const int*   __restrict__ blank_id, // [1]
               int*         __restrict__ map,      // [N, T] (ws)
               int*         __restrict__ lensbuf,  // [N]    (ws)
               float*       __restrict__ out_lens, // [N]    (d_out tail)
               int T, int V) {
    const int n    = blockIdx.x;
    const int lane = threadIdx.x;           // 0..31 (wave32)
    const int len_n = x_lens[n];
    const float* col = ctc + (size_t)n * (size_t)T * (size_t)V + blank_id[0];

    int count = 0;
    for (int t0 = 0; t0 < T; t0 += 32) {
        const int t = t0 + lane;            // T is a multiple of 32
        const float v = col[(size_t)t * (size_t)V];
        const bool keep = (v < LOG_0P9) && (t < len_n);
        const unsigned mask = ballot32(keep);
        if (keep) {
            const int pre = __popc(mask & ((1u << lane) - 1u));
            map[n * T + count + pre] = t;
        }
        count += __popc(mask);
    }
    if (lane == 0) {
        lensbuf[n]  = count;
        out_lens[n] = (float)count;         // lens_fr, flattened after x_fr
    }
}

// ---------------------------------------------------------------------------
// Kernel 2: gather/zero-pad rows of x into the output.
// One block per output row (n, tp); 128 threads x b128 = 2KB row.
// Valid rows are moved via the gfx1250 async global->LDS->global DMA path
// (tracked by ASYNCcnt, drained with s_wait_asynccnt). Padding rows are
// zero-filled with plain stores.
// ---------------------------------------------------------------------------
__global__ void __launch_bounds__(128)
fr_gather_kernel(const float* __restrict__ x,       // [N, T, C]
                 const int*   __restrict__ map,     // [N, T]
                 const int*   __restrict__ lensbuf, // [N]
                 float*       __restrict__ out,     // [N, Tp, C]
                 int T, int C, int Tp) {
    __shared__ float4 buf[128];             // 2KB staging tile

    const int n   = blockIdx.y;
    const int tp  = blockIdx.x;
    const int tid = threadIdx.x;            // 0..127, one b128 each (C=512)

    float4* dst = (float4*)(out + ((size_t)n * (size_t)Tp + (size_t)tp) * (size_t)C) + tid;
    const int len = lensbuf[n];

    if (tp < len) {
        const int src_t = map[n * T + tp];
        const float4* src =
            (const float4*)(x + ((size_t)n * (size_t)T + (size_t)src_t) * (size_t)C) + tid;
#if defined(__HIP_DEVICE_COMPILE__) && \
    __has_builtin(__builtin_amdgcn_global_load_async_to_lds_b128) && \
    __has_builtin(__builtin_amdgcn_global_store_async_from_lds_b128)
        // gfx1250 async DMA path: global -> LDS, drain ASYNCcnt, LDS -> global.
        // Each lane touches only its own LDS slot, so no barrier is needed.
        __builtin_amdgcn_global_load_async_to_lds_b128(
            (g_v4i_ptr)src, (l_v4i_ptr)&buf[tid], 0, 0);
        wait_async_zero();
        __builtin_amdgcn_global_store_async_from_lds_b128(
            (g_v4i_ptr)dst, (l_v4i_ptr)&buf[tid], 0, 0);
        wait_async_zero();
#else
        *dst = *src;                        // fallback: plain b128 copy
        (void)buf;
#endif
    } else {
        *dst = make_float4(0.f, 0.f, 0.f, 0.f);
    }
}

// ---------------------------------------------------------------------------
// Host launcher.
// d_in order: x [N,T,C] f32, x_lens [N] i32, ctc_output [N,T,V] f32, blank_id [1] i32
// d_out: x_fr [N,Tp,C] f32 followed by lens_fr [N] (as f32). Tp recovered from out_size.
// d_ws: int map[N*T] + int lensbuf[N]  (~128KB + 64B).
// ---------------------------------------------------------------------------
extern "C" void kernel_launch(void* const* d_in, const int* in_sizes, int n_in,
                              void* d_out, int out_size, void* d_ws, size_t ws_size,
                              hipStream_t stream) {
    const float* x        = (const float*)d_in[0];
    const int*   x_lens   = (const int*)d_in[1];
    const float* ctc      = (const float*)d_in[2];
    const int*   blank_id = (const int*)d_in[3];

    const int N  = in_sizes[1];
    const int C  = 512;                               // reference constant
    const int T  = in_sizes[0] / (N * C);
    const int V  = in_sizes[2] / (N * T);
    const int Tp = (out_size - N) / (N * C);          // out = N*Tp*C + N

    int* map     = (int*)d_ws;                        // [N*T]
    int* lensbuf = map + (size_t)N * (size_t)T;       // [N]

    float* out      = (float*)d_out;
    float* out_lens = out + (size_t)N * (size_t)Tp * (size_t)C;

    fr_scan_kernel<<<N, 32, 0, stream>>>(ctc, x_lens, blank_id,
                                         map, lensbuf, out_lens, T, V);

    if (Tp > 0) {
        dim3 grid((unsigned)Tp, (unsigned)N);
        fr_gather_kernel<<<grid, 128, 0, stream>>>(x, map, lensbuf, out, T, C, Tp);
    }
}